// Mamba2Block_90486370992207
// MI455X (gfx1250) — compile-verified
//
#include <hip/hip_runtime.h>
#include <hip/hip_bf16.h>

// ---------------- problem constants ----------------
#define SEQ    1024
#define DIM    1024
#define DIN    2048      // d_inner
#define NPROJ  4256      // 2*DIN + 2*DST + NH
#define DST    64        // d_state
#define NH     32        // n_heads
#define HD     64        // head_dim
#define ZOFF   0
#define XOFF   (DIN)            // 2048
#define BOFF   (2*DIN)          // 4096
#define COFF   (2*DIN + DST)    // 4160
#define DTOFF  (2*DIN + 2*DST)  // 4224

typedef __attribute__((ext_vector_type(2))) float f32x2;
typedef __attribute__((ext_vector_type(8))) float f32x8;
typedef __attribute__((__vector_size__(4 * sizeof(int)))) int i32x4;

#define AS1 __attribute__((address_space(1)))
#define AS3 __attribute__((address_space(3)))

#if defined(__has_builtin)
#if __has_builtin(__builtin_amdgcn_global_load_async_to_lds_b128) && \
    __has_builtin(__builtin_amdgcn_s_wait_asynccnt)
#define HAVE_ASYNC_LDS 1
#endif
#endif

// ---------------------------------------------------------------------------
// GEMM:  C[m,n] = sum_k A[m,k] * B[n,k]   (A: MxK row-major, B: NxK row-major)
// fp32 WMMA 16x16x4. Wave computes a 32(M) x 64(N) tile (8 accumulators).
// Block = 8 waves arranged 4(M) x 2(N)  => 128 x 128 block tile.
// N-edge handled by clamping B row pointers; K loop is branch-free.
// ---------------------------------------------------------------------------
#define WMMA_F32(a, b, c) \
  __builtin_amdgcn_wmma_f32_16x16x4_f32(false, (a), false, (b), (short)0, (c), false, false)

__global__ __launch_bounds__(256)
void gemm_nt_wmma_f32(const float* __restrict__ A, const float* __restrict__ B,
                      float* __restrict__ C, int M, int N, int K) {
  const int lane = threadIdx.x & 31;
  const int wave = threadIdx.x >> 5;       // 0..7
  const int wm   = wave >> 1;              // 0..3 (M direction)
  const int wn   = wave & 1;               // 0..1 (N direction)
  const int m0   = blockIdx.y * 128 + wm * 32;
  const int n0   = blockIdx.x * 128 + wn * 64;
  const int l16  = lane & 15;
  const int ak   = (lane >> 4) * 2;        // K-pair select

  const float* __restrict__ a0 = A + (size_t)(m0 + l16) * K;
  const float* __restrict__ a1 = a0 + (size_t)16 * K;

  const bool sv0 = (n0 +  16) <= N;
  const bool sv1 = (n0 +  32) <= N;
  const bool sv2 = (n0 +  48) <= N;
  const bool sv3 = (n0 +  64) <= N;
  int bn0 = n0 +  0 + l16; if (bn0 >= N) bn0 = N - 1;
  int bn1 = n0 + 16 + l16; if (bn1 >= N) bn1 = N - 1;
  int bn2 = n0 + 32 + l16; if (bn2 >= N) bn2 = N - 1;
  int bn3 = n0 + 48 + l16; if (bn3 >= N) bn3 = N - 1;
  const float* __restrict__ b0 = B + (size_t)bn0 * K;
  const float* __restrict__ b1 = B + (size_t)bn1 * K;
  const float* __restrict__ b2 = B + (size_t)bn2 * K;
  const float* __restrict__ b3 = B + (size_t)bn3 * K;

  f32x8 acc00 = {}, acc01 = {}, acc02 = {}, acc03 = {};
  f32x8 acc10 = {}, acc11 = {}, acc12 = {}, acc13 = {};

  for (int k = 0; k < K; k += 8) {
    f32x2 afa0 = *(const f32x2*)(a0 + k + ak);
    f32x2 afa1 = *(const f32x2*)(a1 + k + ak);
    f32x2 bfa0 = *(const f32x2*)(b0 + k + ak);
    f32x2 bfa1 = *(const f32x2*)(b1 + k + ak);
    f32x2 bfa2 = *(const f32x2*)(b2 + k + ak);
    f32x2 bfa3 = *(const f32x2*)(b3 + k + ak);
    f32x2 afb0 = *(const f32x2*)(a0 + k + 4 + ak);
    f32x2 afb1 = *(const f32x2*)(a1 + k + 4 + ak);
    f32x2 bfb0 = *(const f32x2*)(b0 + k + 4 + ak);
    f32x2 bfb1 = *(const f32x2*)(b1 + k + 4 + ak);
    f32x2 bfb2 = *(const f32x2*)(b2 + k + 4 + ak);
    f32x2 bfb3 = *(const f32x2*)(b3 + k + 4 + ak);

    acc00 = WMMA_F32(afa0, bfa0, acc00);
    acc01 = WMMA_F32(afa0, bfa1, acc01);
    acc02 = WMMA_F32(afa0, bfa2, acc02);
    acc03 = WMMA_F32(afa0, bfa3, acc03);
    acc10 = WMMA_F32(afa1, bfa0, acc10);
    acc11 = WMMA_F32(afa1, bfa1, acc11);
    acc12 = WMMA_F32(afa1, bfa2, acc12);
    acc13 = WMMA_F32(afa1, bfa3, acc13);

    acc00 = WMMA_F32(afb0, bfb0, acc00);
    acc01 = WMMA_F32(afb0, bfb1, acc01);
    acc02 = WMMA_F32(afb0, bfb2, acc02);
    acc03 = WMMA_F32(afb0, bfb3, acc03);
    acc10 = WMMA_F32(afb1, bfb0, acc10);
    acc11 = WMMA_F32(afb1, bfb1, acc11);
    acc12 = WMMA_F32(afb1, bfb2, acc12);
    acc13 = WMMA_F32(afb1, bfb3, acc13);
  }

  const int rb = (lane >> 4) * 8;
#pragma unroll
  for (int r = 0; r < 8; ++r) {
    float* crow0 = C + (size_t)(m0 +      rb + r) * N;
    float* crow1 = C + (size_t)(m0 + 16 + rb + r) * N;
    if (sv0) { crow0[n0 +  0 + l16] = acc00[r]; crow1[n0 +  0 + l16] = acc10[r]; }
    if (sv1) { crow0[n0 + 16 + l16] = acc01[r]; crow1[n0 + 16 + l16] = acc11[r]; }
    if (sv2) { crow0[n0 + 32 + l16] = acc02[r]; crow1[n0 + 32 + l16] = acc12[r]; }
    if (sv3) { crow0[n0 + 48 + l16] = acc03[r]; crow1[n0 + 48 + l16] = acc13[r]; }
  }
}

// ---------------------------------------------------------------------------
// Causal depthwise conv1d (width 4) + bias + SiLU over the xin slice of P.
// ---------------------------------------------------------------------------
__global__ __launch_bounds__(256)
void conv_silu_kernel(const float* __restrict__ P, const float* __restrict__ Wc,
                      const float* __restrict__ bc, float* __restrict__ xc) {
  int idx = blockIdx.x * blockDim.x + threadIdx.x;
  if (idx >= SEQ * DIN) return;
  int t = idx >> 11;            // /2048
  int c = idx & (DIN - 1);
  float acc = bc[c];
#pragma unroll
  for (int j = 0; j < 4; ++j) {
    int tau = t - 3 + j;
    if (tau >= 0) acc = fmaf(P[(size_t)tau * NPROJ + XOFF + c], Wc[c * 4 + j], acc);
  }
  xc[idx] = acc / (1.0f + __expf(-acc));   // SiLU
}

// softplus(dt + dt_bias) -> dtsp[t*NH + h]
__global__ __launch_bounds__(256)
void dt_softplus_kernel(const float* __restrict__ P, const float* __restrict__ dtb,
                        float* __restrict__ dtsp) {
  int idx = blockIdx.x * blockDim.x + threadIdx.x;
  if (idx >= SEQ * NH) return;
  int t = idx >> 5;
  int h = idx & (NH - 1);
  float v = P[(size_t)t * NPROJ + DTOFF + h] + dtb[h];
  dtsp[idx] = (v > 20.0f) ? v : log1pf(__expf(v));
}

// ---------------------------------------------------------------------------
// Selective-state scan. One block per head (32 blocks x 512 threads = 16
// wave32 waves). thread = (d, nb): owns 8 states in registers.
//
// Per timestep, the whole block shares B_t (256B), C_t (256B) and the head's
// x_t slice (256B). These are staged into a double-buffered LDS tile one step
// ahead via GLOBAL_LOAD_ASYNC_TO_LDS_B128 (tracked by ASYNCcnt, drained with
// s_wait_asynccnt 0), so the per-thread inner loop reads ds_load_b128 instead
// of 32KB/step of redundant VMEM. global_prefetch covers the B/C row stream.
// ---------------------------------------------------------------------------
__global__ __launch_bounds__(512)
void ssm_scan_kernel(const float* __restrict__ P, const float* __restrict__ xc,
                     const float* __restrict__ dtsp, const float* __restrict__ A_log,
                     float* __restrict__ yh) {
  const int h   = blockIdx.x;          // head
  const int tid = threadIdx.x;
  const int d   = tid >> 3;            // 0..63
  const int nb  = tid & 7;             // 0..7
  const int n0  = nb * 8;

  __shared__ __align__(16) float sBC[2][2 * DST];  // [buf][B(64) | C(64)]
  __shared__ __align__(16) float sX[2][HD];        // [buf][x slice]

  // stage step t into buffer (t&1): threads 0..31 copy B|C (512B contiguous),
  // threads 32..47 copy the x_t slice (256B).
  auto stage = [&](int t) {
    const int buf = t & 1;
    if (tid < 32) {
      const float* gsrc = P + (size_t)t * NPROJ + BOFF + tid * 4;
#ifdef HAVE_ASYNC_LDS
      __builtin_amdgcn_global_load_async_to_lds_b128(
          (AS1 i32x4*)gsrc, (AS3 i32x4*)&sBC[buf][tid * 4], 0, 0);
#else
      *(float4*)&sBC[buf][tid * 4] = *(const float4*)gsrc;
#endif
    } else if (tid < 48) {
      const int i = tid - 32;
      const float* gsrc = xc + (size_t)t * DIN + h * HD + i * 4;
#ifdef HAVE_ASYNC_LDS
      __builtin_amdgcn_global_load_async_to_lds_b128(
          (AS1 i32x4*)gsrc, (AS3 i32x4*)&sX[buf][i * 4], 0, 0);
#else
      *(float4*)&sX[buf][i * 4] = *(const float4*)gsrc;
#endif
    }
  };

  // A = -exp(A_log[h,d,n])  (constant over t)
  float acoef[8];
#pragma unroll
  for (int j = 0; j < 8; ++j)
    acoef[j] = -__expf(A_log[(size_t)h * HD * DST + (size_t)d * DST + n0 + j]);

  float st[8];
#pragma unroll
  for (int j = 0; j < 8; ++j) st[j] = 0.0f;

  stage(0);
#ifdef HAVE_ASYNC_LDS
  __builtin_amdgcn_s_wait_asynccnt(0);
#endif
  __syncthreads();

  for (int t = 0; t < SEQ; ++t) {
    const int buf = t & 1;
    if (t + 1 < SEQ) stage(t + 1);
    if (tid == 0)  // stream prefetch of the B/C row 4 steps ahead
      __builtin_prefetch(P + (size_t)(t + 4) * NPROJ + BOFF, 0, 1);

    const float dt_t = dtsp[t * NH + h];
    const float x_t  = sX[buf][d];

    float Bv[8], Cv[8];
    *(float4*)&Bv[0] = *(const float4*)&sBC[buf][n0];
    *(float4*)&Bv[4] = *(const float4*)&sBC[buf][n0 + 4];
    *(float4*)&Cv[0] = *(const float4*)&sBC[buf][DST + n0];
    *(float4*)&Cv[4] = *(const float4*)&sBC[buf][DST + n0 + 4];

    float part = 0.0f;
#pragma unroll
    for (int j = 0; j < 8; ++j) {
      float a = __expf(dt_t * acoef[j]);            // exp(dt*A)
      st[j] = fmaf(a, st[j], Bv[j] * x_t);          // h = a*h + B*x
      part  = fmaf(st[j], Cv[j], part);             // y partial = h . C
    }
    part += __shfl_xor(part, 1);
    part += __shfl_xor(part, 2);
    part += __shfl_xor(part, 4);
    if (nb == 0) yh[(size_t)t * DIN + h * HD + d] = part;

#ifdef HAVE_ASYNC_LDS
    __builtin_amdgcn_s_wait_asynccnt(0);
#endif
    __syncthreads();
  }
}

// ---------------------------------------------------------------------------
// y = (yh + xc*D) * silu(z); RMSNorm over d_inner; * norm_w  -> g
// ---------------------------------------------------------------------------
__global__ __launch_bounds__(256)
void gate_rmsnorm_kernel(const float* __restrict__ P, const float* __restrict__ yh,
                         const float* __restrict__ xc, const float* __restrict__ Dp,
                         const float* __restrict__ nw, float* __restrict__ g) {
  const int t   = blockIdx.x;
  const int tid = threadIdx.x;
  const float* __restrict__ prow = P + (size_t)t * NPROJ;

  float vals[8];
  float ss = 0.0f;
#pragma unroll
  for (int i = 0; i < 8; ++i) {
    int c = tid + i * 256;
    float y = fmaf(xc[(size_t)t * DIN + c], Dp[c], yh[(size_t)t * DIN + c]);
    float z = prow[ZOFF + c];
    y *= z / (1.0f + __expf(-z));    // gate with silu(z)
    vals[i] = y;
    ss = fmaf(y, y, ss);
  }
#pragma unroll
  for (int off = 16; off > 0; off >>= 1) ss += __shfl_xor(ss, off);
  __shared__ float red[8];
  if ((tid & 31) == 0) red[tid >> 5] = ss;
  __syncthreads();
  float tot = 0.0f;
#pragma unroll
  for (int w = 0; w < 8; ++w) tot += red[w];
  const float scale = rsqrtf(tot * (1.0f / DIN) + 1e-6f);
#pragma unroll
  for (int i = 0; i < 8; ++i) {
    int c = tid + i * 256;
    g[(size_t)t * DIN + c] = vals[i] * scale * nw[c];
  }
}

// ---------------------------------------------------------------------------
extern "C" void kernel_launch(void* const* d_in, const int* in_sizes, int n_in,
                              void* d_out, int out_size, void* d_ws, size_t ws_size,
                              hipStream_t stream) {
  const float* x      = (const float*)d_in[0];  // (1,1024,1024)
  const float* W_in   = (const float*)d_in[1];  // (4256,1024)
  const float* W_conv = (const float*)d_in[2];  // (2048,1,4)
  const float* b_conv = (const float*)d_in[3];  // (2048)
  const float* A_log  = (const float*)d_in[4];  // (32,64,64)
  const float* Dp     = (const float*)d_in[5];  // (32,64) -> flat 2048
  const float* dt_b   = (const float*)d_in[6];  // (32)
  const float* norm_w = (const float*)d_in[7];  // (2048)
  const float* W_out  = (const float*)d_in[8];  // (1024,2048)
  float* out = (float*)d_out;                   // (1,1024,1024)

  // workspace layout (floats)
  float* P    = (float*)d_ws;                        // SEQ x NPROJ
  float* xc   = P    + (size_t)SEQ * NPROJ;          // SEQ x DIN
  float* dtsp = xc   + (size_t)SEQ * DIN;            // SEQ x NH
  float* yh   = dtsp + (size_t)SEQ * NH;             // SEQ x DIN
  float* g    = yh   + (size_t)SEQ * DIN;            // SEQ x DIN

  // 1) in-projection: P = x @ W_in^T   (M=1024, N=4256, K=1024)
  gemm_nt_wmma_f32<<<dim3((NPROJ + 127) / 128, SEQ / 128), 256, 0, stream>>>(
      x, W_in, P, SEQ, NPROJ, DIM);

  // 2) depthwise causal conv + SiLU; softplus(dt)
  conv_silu_kernel<<<(SEQ * DIN + 255) / 256, 256, 0, stream>>>(P, W_conv, b_conv, xc);
  dt_softplus_kernel<<<(SEQ * NH + 255) / 256, 256, 0, stream>>>(P, dt_b, dtsp);

  // 3) sequential selective scan, parallel over (h, d, n)
  ssm_scan_kernel<<<NH, 512, 0, stream>>>(P, xc, dtsp, A_log, yh);

  // 4) skip + gate + RMSNorm
  gate_rmsnorm_kernel<<<SEQ, 256, 0, stream>>>(P, yh, xc, Dp, norm_w, g);

  // 5) out-projection: out = g @ W_out^T (M=1024, N=1024, K=2048)
  gemm_nt_wmma_f32<<<dim3(DIM / 128, SEQ / 128), 256, 0, stream>>>(
      g, W_out, out, SEQ, DIM, DIN);
}